// TMPHN_12128987644192
// MI455X (gfx1250) — compile-verified
//
#include <hip/hip_runtime.h>
#include <math.h>

// CDNA5 / gfx1250: wave32, WMMA 16x16 tiles.
typedef __attribute__((ext_vector_type(2))) float v2f;
typedef __attribute__((ext_vector_type(8))) float v8f;

#define NPB 64          // nodes per block
#define ASTRIDE 68      // 64 + 4 pad: bank = (4*row + col) % 64 -> conflict-free A-frag b64 LDS reads

__device__ __forceinline__ void acc4(float4& a, const float4 b) {
    a.x += b.x; a.y += b.y; a.z += b.z; a.w += b.w;
}

// One fused layer: masked-mean neighbor aggregation (gather) -> [tile,64] GEMM (f32 WMMA) -> ReLU.
// FINAL additionally applies the 64x40 classifier + bias + log_softmax.
template <bool FINAL>
__global__ __launch_bounds__(256)
void hgnn_layer(const float* __restrict__ src,   // [N,64] node features (X or h1)
                const int*   __restrict__ nidx,  // [N,32] neighbor indices
                const int*   __restrict__ deg,   // [N]
                const float* __restrict__ W,     // [64,64] layer weight (W1 or W2)
                const float* __restrict__ Wc,    // [64,40] classifier weight (FINAL)
                const float* __restrict__ bias,  // [40]    classifier bias   (FINAL)
                float* __restrict__ out,         // !FINAL: h [N,64]; FINAL: log_softmax [N,40]
                int N)
{
    __shared__ float agg[NPB * ASTRIDE];   // aggregated features / (FINAL) h2 tile
    __shared__ v2f   wpack[32 * 64];       // W pre-packed as B-fragments: {W[2k][n], W[2k+1][n]}
    __shared__ float wcls[64 * 40];        // FINAL: Wc staged in LDS
    __shared__ float bsh[40];

    const int tid = threadIdx.x;

    // ---- phase 0: stage weights into LDS --------------------------------
    for (int i = tid; i < 32 * 64; i += 256) {
        const int kk = i >> 6, n = i & 63;
        v2f w = { W[(2 * kk) * 64 + n], W[(2 * kk + 1) * 64 + n] };
        wpack[i] = w;
    }
    if (FINAL) {
        for (int i = tid; i < 64 * 40; i += 256) wcls[i] = Wc[i];
        if (tid < 40) bsh[tid] = bias[tid];
    }

    // ---- phase 1: masked-mean aggregation (irregular gather, L2-bound) --
    {
        const int nl = tid >> 2;            // local node 0..63
        const int sub = tid & 3;            // 16-column slice owner
        const int node = blockIdx.x * NPB + nl;
        if (node < N) {
            int dc = deg[node]; if (dc < 1) dc = 1;
            const float invd = 1.0f / (float)dc;
            const int* __restrict__ nrow = nidx + (size_t)node * 32;
            float4 a0 = {0.f, 0.f, 0.f, 0.f}, a1 = a0, a2 = a0, a3 = a0;
            for (int j = 0; j < dc; ++j) {
                const int nb = nrow[j];
                const float4* __restrict__ r =
                    (const float4*)(src + (size_t)nb * 64) + sub * 4;
                acc4(a0, r[0]); acc4(a1, r[1]); acc4(a2, r[2]); acc4(a3, r[3]);
            }
            float4* arow = (float4*)(&agg[nl * ASTRIDE + sub * 16]);
            arow[0] = make_float4(a0.x * invd, a0.y * invd, a0.z * invd, a0.w * invd);
            arow[1] = make_float4(a1.x * invd, a1.y * invd, a1.z * invd, a1.w * invd);
            arow[2] = make_float4(a2.x * invd, a2.y * invd, a2.z * invd, a2.w * invd);
            arow[3] = make_float4(a3.x * invd, a3.y * invd, a3.z * invd, a3.w * invd);
        }
    }
    __syncthreads();

    // ---- phase 2: [64,64] @ [64,64] via V_WMMA_F32_16X16X4_F32 ----------
    // 8 waves: wave w -> M-tile (w>>1)*16, N-half (w&1)*32 (two 16x16 tiles).
    const int wv = tid >> 5, ln = tid & 31;
    const int m0 = (wv >> 1) * 16;
    const int n0 = (wv & 1) * 32;
    const int lr = ln & 15;                 // row within tile (A) / col (B,D)
    const int kh = ln >> 4;                 // half-wave selector
    const float* arow = &agg[(m0 + lr) * ASTRIDE];

    v8f c0 = {0.f, 0.f, 0.f, 0.f, 0.f, 0.f, 0.f, 0.f};
    v8f c1 = c0;
#pragma unroll
    for (int kk4 = 0; kk4 < 16; ++kk4) {
        const int k = kk4 * 4;
        // A frag (16x4 f32): V0 = K=k (lanes0-15) / k+2 (lanes16-31); V1 = k+1 / k+3
        v2f a = *(const v2f*)(arow + k + 2 * kh);
        // B frag (4x16 f32): lane = col; V0/V1 = rows {k,k+1} (lo half) / {k+2,k+3} (hi half)
        v2f b0 = wpack[(2 * kk4 + kh) * 64 + n0 + lr];
        v2f b1 = wpack[(2 * kk4 + kh) * 64 + n0 + 16 + lr];
        c0 = __builtin_amdgcn_wmma_f32_16x16x4_f32(false, a, false, b0, (short)0, c0, false, false);
        c1 = __builtin_amdgcn_wmma_f32_16x16x4_f32(false, a, false, b1, (short)0, c1, false, false);
    }

    if (!FINAL) {
        // D layout: VGPR r, lanes0-15 -> M=r, lanes16-31 -> M=r+8; N = lane&15
#pragma unroll
        for (int r = 0; r < 8; ++r) {
            const int node = blockIdx.x * NPB + m0 + 8 * kh + r;
            if (node < N) {
                out[(size_t)node * 64 + n0 + lr]      = fmaxf(c0[r], 0.0f);
                out[(size_t)node * 64 + n0 + 16 + lr] = fmaxf(c1[r], 0.0f);
            }
        }
    } else {
        // write ReLU(h2) back into the LDS tile (barrier: other wave still reads these rows)
        __syncthreads();
#pragma unroll
        for (int r = 0; r < 8; ++r) {
            const int row = m0 + 8 * kh + r;
            agg[row * ASTRIDE + n0 + lr]      = fmaxf(c0[r], 0.0f);
            agg[row * ASTRIDE + n0 + 16 + lr] = fmaxf(c1[r], 0.0f);
        }
        __syncthreads();

        // ---- phase 3: classifier + log_softmax --------------------------
        // 4 lanes per node (aligned group within wave32), 10 classes each.
        const int nl = tid >> 2, sub = tid & 3;
        const int node = blockIdx.x * NPB + nl;
        if (node < N) {
            float lg[10];
#pragma unroll
            for (int c = 0; c < 10; ++c) lg[c] = bsh[sub * 10 + c];
            const float* hrow = &agg[nl * ASTRIDE];
            for (int k = 0; k < 64; ++k) {
                const float h = hrow[k];
                const float* wr = &wcls[k * 40 + sub * 10];
#pragma unroll
                for (int c = 0; c < 10; ++c) lg[c] = fmaf(h, wr[c], lg[c]);
            }
            float mx = lg[0];
#pragma unroll
            for (int c = 1; c < 10; ++c) mx = fmaxf(mx, lg[c]);
            mx = fmaxf(mx, __shfl_xor(mx, 1, 32));
            mx = fmaxf(mx, __shfl_xor(mx, 2, 32));
            float s = 0.0f;
#pragma unroll
            for (int c = 0; c < 10; ++c) s += expf(lg[c] - mx);
            s += __shfl_xor(s, 1, 32);
            s += __shfl_xor(s, 2, 32);
            const float lse = mx + logf(s);
            float* orow = out + (size_t)node * 40 + sub * 10;
#pragma unroll
            for (int c = 0; c < 10; ++c) orow[c] = lg[c] - lse;
        }
    }
}

extern "C" void kernel_launch(void* const* d_in, const int* in_sizes, int n_in,
                              void* d_out, int out_size, void* d_ws, size_t ws_size,
                              hipStream_t stream) {
    const float* X    = (const float*)d_in[0];
    const int*   nidx = (const int*)d_in[1];
    const int*   deg  = (const int*)d_in[2];
    // d_in[3] = nodes (arange identity, unused)
    const float* W1   = (const float*)d_in[4];
    const float* W2   = (const float*)d_in[5];
    const float* Wc   = (const float*)d_in[6];
    const float* b    = (const float*)d_in[7];
    const int N = in_sizes[2];                 // deg has N elements

    float* h1 = (float*)d_ws;                  // needs N*64*4 = 25.6 MB of workspace
    float* outp = (float*)d_out;

    const int blocks = (N + NPB - 1) / NPB;
    hgnn_layer<false><<<blocks, 256, 0, stream>>>(X,  nidx, deg, W1, Wc, b, h1,   N);
    hgnn_layer<true ><<<blocks, 256, 0, stream>>>(h1, nidx, deg, W2, Wc, b, outp, N);
}